// DecoupledTopKGate_75024488726885
// MI455X (gfx1250) — compile-verified
//
#include <hip/hip_runtime.h>
#include <math.h>

typedef __attribute__((ext_vector_type(2))) float v2f;
typedef __attribute__((ext_vector_type(4))) float v4f;
typedef __attribute__((ext_vector_type(8))) float v8f;

#define TOKENS      8192
#define MODEL_DIM   4096
#define NUM_EXPERTS 64
#define KC          32          // K-chunk staged in LDS
#define KCP         (KC + 2)    // padded column stride -> conflict-free ds reads

// one async b128 copy: global (src+OFF bytes) -> LDS (dst+OFF bytes)
#define ASYNC_CP_B128(dst, src, OFF)                                          \
    asm volatile("global_load_async_to_lds_b128 %0, %1, off offset:" #OFF     \
                 :: "v"(dst), "v"((const void*)(src)) : "memory")

// ordered LDS fragment-pair load: q = { frag(kstep 2j), frag(kstep 2j+1) }
#define DS_LOAD_PAIR(q, addr, O0, O1)                                         \
    asm volatile("ds_load_2addr_b64 %0, %1 offset0:" #O0 " offset1:" #O1      \
                 : "=v"(q) : "v"(addr) : "memory")

// -------------------------------------------------------------------------
// Kernel 1: fused dual gating GEMM (f32 WMMA 16x16x4).
//   block = 256 threads = 8 waves; each wave: 16 tokens x 128 cols
//   cols 0..63   -> selection_scores (w_sel)
//   cols 64..127 -> fusion_scores    (w_fus), scaled by momentum_weights[e]
//   B panel double-buffered in LDS via async global->LDS DMA; B fragments
//   loaded in ordered groups of 8 (asm) so one dscnt wait covers 16 WMMAs.
// -------------------------------------------------------------------------
__global__ __launch_bounds__(256)
void gate_gemm_kernel(const float* __restrict__ x,
                      const float* __restrict__ w_sel,
                      const float* __restrict__ w_fus,
                      const float* __restrict__ mw,
                      float* __restrict__ sel_out,
                      float* __restrict__ fus_out)
{
    __shared__ float lds_b[2][128 * KCP];      // 2 x 17 KB weight panels

    const int lane = threadIdx.x & 31;
    const int wave = threadIdx.x >> 5;
    const int token_base = blockIdx.x * 128 + wave * 16;
    const int m  = lane & 15;                  // M (A) / N (B,D) position
    const int kh = lane >> 4;                  // K half: 0 -> K{0,1}, 1 -> K{2,3}

    v8f acc[8] = {};                           // 8 tiles of 16x16 f32

    const float* a_base = x + (size_t)(token_base + m) * MODEL_DIM + kh * 2;

    // staging assignment: thread -> 64B (16 floats) of one weight row
    const int s_col = threadIdx.x >> 1;                    // 0..127
    const int s_k   = (threadIdx.x & 1) * 16;              // float 0 or 16
    const float* s_src =
        ((s_col < 64) ? w_sel : w_fus) + (size_t)(s_col & 63) * MODEL_DIM + s_k;

    auto stage = [&](int buf, int kc) {
        const float* src = s_src + kc;
        unsigned dst = (unsigned)(uintptr_t)&lds_b[buf][s_col * KCP + s_k];
        ASYNC_CP_B128(dst, src, 0);
        ASYNC_CP_B128(dst, src, 16);
        ASYNC_CP_B128(dst, src, 32);
        ASYNC_CP_B128(dst, src, 48);
    };

    const int nchunks = MODEL_DIM / KC;        // 128
    stage(0, 0);

    for (int c = 0; c < nchunks; ++c) {
        const int buf = c & 1;
        if (c + 1 < nchunks) {
            stage(buf ^ 1, (c + 1) * KC);      // prefetch next panel
            asm volatile("s_wait_asynccnt 0x4" ::: "memory");  // chunk c done
        } else {
            asm volatile("s_wait_asynccnt 0x0" ::: "memory");
        }
        __syncthreads();

        const float* a_ptr = a_base + c * KC;
        // per-tile LDS byte addresses (tile stride = 16 cols * KCP floats)
        const unsigned ba =
            (unsigned)(uintptr_t)&lds_b[buf][m * KCP + kh * 2];

        // j covers k-steps {2j, 2j+1}; offsets in b64 units: 4j and 4j+2
#define GEMM_STEP(J, O0, O1)                                                  \
        {                                                                     \
            v4f q0, q1, q2, q3, q4, q5, q6, q7;                               \
            DS_LOAD_PAIR(q0, ba + 0 * 16 * KCP * 4, O0, O1);                  \
            DS_LOAD_PAIR(q1, ba + 1 * 16 * KCP * 4, O0, O1);                  \
            DS_LOAD_PAIR(q2, ba + 2 * 16 * KCP * 4, O0, O1);                  \
            DS_LOAD_PAIR(q3, ba + 3 * 16 * KCP * 4, O0, O1);                  \
            DS_LOAD_PAIR(q4, ba + 4 * 16 * KCP * 4, O0, O1);                  \
            DS_LOAD_PAIR(q5, ba + 5 * 16 * KCP * 4, O0, O1);                  \
            DS_LOAD_PAIR(q6, ba + 6 * 16 * KCP * 4, O0, O1);                  \
            DS_LOAD_PAIR(q7, ba + 7 * 16 * KCP * 4, O0, O1);                  \
            v2f a0 = *(const v2f*)(a_ptr + 8 * (J));                          \
            v2f a1 = *(const v2f*)(a_ptr + 8 * (J) + 4);                      \
            acc[0] = __builtin_amdgcn_wmma_f32_16x16x4_f32(false, a0, false,  \
                __builtin_shufflevector(q0, q0, 0, 1), (short)0, acc[0], false, false); \
            acc[1] = __builtin_amdgcn_wmma_f32_16x16x4_f32(false, a0, false,  \
                __builtin_shufflevector(q1, q1, 0, 1), (short)0, acc[1], false, false); \
            acc[2] = __builtin_amdgcn_wmma_f32_16x16x4_f32(false, a0, false,  \
                __builtin_shufflevector(q2, q2, 0, 1), (short)0, acc[2], false, false); \
            acc[3] = __builtin_amdgcn_wmma_f32_16x16x4_f32(false, a0, false,  \
                __builtin_shufflevector(q3, q3, 0, 1), (short)0, acc[3], false, false); \
            acc[4] = __builtin_amdgcn_wmma_f32_16x16x4_f32(false, a0, false,  \
                __builtin_shufflevector(q4, q4, 0, 1), (short)0, acc[4], false, false); \
            acc[5] = __builtin_amdgcn_wmma_f32_16x16x4_f32(false, a0, false,  \
                __builtin_shufflevector(q5, q5, 0, 1), (short)0, acc[5], false, false); \
            acc[6] = __builtin_amdgcn_wmma_f32_16x16x4_f32(false, a0, false,  \
                __builtin_shufflevector(q6, q6, 0, 1), (short)0, acc[6], false, false); \
            acc[7] = __builtin_amdgcn_wmma_f32_16x16x4_f32(false, a0, false,  \
                __builtin_shufflevector(q7, q7, 0, 1), (short)0, acc[7], false, false); \
            acc[0] = __builtin_amdgcn_wmma_f32_16x16x4_f32(false, a1, false,  \
                __builtin_shufflevector(q0, q0, 2, 3), (short)0, acc[0], false, false); \
            acc[1] = __builtin_amdgcn_wmma_f32_16x16x4_f32(false, a1, false,  \
                __builtin_shufflevector(q1, q1, 2, 3), (short)0, acc[1], false, false); \
            acc[2] = __builtin_amdgcn_wmma_f32_16x16x4_f32(false, a1, false,  \
                __builtin_shufflevector(q2, q2, 2, 3), (short)0, acc[2], false, false); \
            acc[3] = __builtin_amdgcn_wmma_f32_16x16x4_f32(false, a1, false,  \
                __builtin_shufflevector(q3, q3, 2, 3), (short)0, acc[3], false, false); \
            acc[4] = __builtin_amdgcn_wmma_f32_16x16x4_f32(false, a1, false,  \
                __builtin_shufflevector(q4, q4, 2, 3), (short)0, acc[4], false, false); \
            acc[5] = __builtin_amdgcn_wmma_f32_16x16x4_f32(false, a1, false,  \
                __builtin_shufflevector(q5, q5, 2, 3), (short)0, acc[5], false, false); \
            acc[6] = __builtin_amdgcn_wmma_f32_16x16x4_f32(false, a1, false,  \
                __builtin_shufflevector(q6, q6, 2, 3), (short)0, acc[6], false, false); \
            acc[7] = __builtin_amdgcn_wmma_f32_16x16x4_f32(false, a1, false,  \
                __builtin_shufflevector(q7, q7, 2, 3), (short)0, acc[7], false, false); \
        }

        GEMM_STEP(0, 0, 2)
        GEMM_STEP(1, 4, 6)
        GEMM_STEP(2, 8, 10)
        GEMM_STEP(3, 12, 14)
#undef GEMM_STEP

        __syncthreads();
    }

    // --- epilogue: D layout: VGPR r -> M = r + 8*kh, N = lane&15 ---
    const int token0 = token_base + kh * 8;
#pragma unroll
    for (int t = 0; t < 8; ++t) {
        if (t < 4) {
            int e = t * 16 + m;
#pragma unroll
            for (int r = 0; r < 8; ++r)
                sel_out[(size_t)(token0 + r) * NUM_EXPERTS + e] = acc[t][r];
        } else {
            int e = (t - 4) * 16 + m;
            float s = mw[e];                   // pre-update momentum weight
#pragma unroll
            for (int r = 0; r < 8; ++r)
                fus_out[(size_t)(token0 + r) * NUM_EXPERTS + e] = acc[t][r] * s;
        }
    }
}

// -------------------------------------------------------------------------
// Kernel 2: top-2 per token (strict > => lowest index wins ties, like top_k)
// -------------------------------------------------------------------------
__global__ __launch_bounds__(256)
void topk_kernel(const float* __restrict__ sel,
                 float* __restrict__ tv, float* __restrict__ ti)
{
    int t = blockIdx.x * blockDim.x + threadIdx.x;
    if (t >= TOKENS) return;
    const v4f* row = (const v4f*)(sel + (size_t)t * NUM_EXPERTS);
    float v1 = -INFINITY, v2 = -INFINITY;
    int   i1 = 0,        i2 = 0;
#pragma unroll
    for (int i = 0; i < NUM_EXPERTS / 4; ++i) {
        v4f q = row[i];
#pragma unroll
        for (int j = 0; j < 4; ++j) {
            float v = q[j];
            int  id = i * 4 + j;
            if (v > v1)      { v2 = v1; i2 = i1; v1 = v; i1 = id; }
            else if (v > v2) { v2 = v;  i2 = id; }
        }
    }
    tv[t * 2 + 0] = v1;
    tv[t * 2 + 1] = v2;
    ti[t * 2 + 0] = (float)i1;
    ti[t * 2 + 1] = (float)i2;
}

// -------------------------------------------------------------------------
// Kernel 3: usage histogram + EMA + inverse-usage softmax (single block)
// -------------------------------------------------------------------------
__global__ __launch_bounds__(256)
void stats_kernel(const float* __restrict__ ti,
                  const float* __restrict__ old_usage,
                  float* __restrict__ new_mw,
                  float* __restrict__ new_usage_out)
{
    __shared__ float hist[NUM_EXPERTS];
    __shared__ float nu_s[NUM_EXPERTS];
    __shared__ float inv_s[NUM_EXPERTS];

    if (threadIdx.x < NUM_EXPERTS) hist[threadIdx.x] = 0.0f;
    __syncthreads();

    for (int i = threadIdx.x; i < TOKENS * 2; i += blockDim.x) {
        int e = (int)ti[i];
        atomicAdd(&hist[e], 1.0f);
    }
    __syncthreads();

    if (threadIdx.x == 0) {
        float sum = 0.0f;
        for (int e = 0; e < NUM_EXPERTS; ++e) {
            nu_s[e] = 0.9f * old_usage[e] + 0.1f * hist[e];
            sum += nu_s[e];
        }
        float mx = -INFINITY;
        for (int e = 0; e < NUM_EXPERTS; ++e) {
            float un = nu_s[e] / (sum + 1e-8f);
            inv_s[e] = 1.0f / (un + 1e-8f);     // TEMPERATURE == 1.0
            mx = fmaxf(mx, inv_s[e]);
        }
        float es = 0.0f;
        for (int e = 0; e < NUM_EXPERTS; ++e) {
            inv_s[e] = __expf(inv_s[e] - mx);
            es += inv_s[e];
        }
        for (int e = 0; e < NUM_EXPERTS; ++e) {
            new_mw[e]        = inv_s[e] / es;
            new_usage_out[e] = nu_s[e];
        }
    }
}

// -------------------------------------------------------------------------
// Launcher
// -------------------------------------------------------------------------
extern "C" void kernel_launch(void* const* d_in, const int* in_sizes, int n_in,
                              void* d_out, int out_size, void* d_ws, size_t ws_size,
                              hipStream_t stream) {
    const float* x      = (const float*)d_in[0];
    const float* w_sel  = (const float*)d_in[1];
    const float* w_fus  = (const float*)d_in[2];
    const float* mw     = (const float*)d_in[3];
    const float* usage  = (const float*)d_in[4];

    float* out = (float*)d_out;
    float* sel_out   = out;                                    // [8192,64]
    float* fus_out   = out + (size_t)TOKENS * NUM_EXPERTS;     // [8192,64]
    float* topk_vals = fus_out + (size_t)TOKENS * NUM_EXPERTS; // [8192,2]
    float* topk_idx  = topk_vals + (size_t)TOKENS * 2;         // [8192,2]
    float* new_mw    = topk_idx + (size_t)TOKENS * 2;          // [64]
    float* new_usage = new_mw + NUM_EXPERTS;                   // [64]

    gate_gemm_kernel<<<TOKENS / 128, 256, 0, stream>>>(
        x, w_sel, w_fus, mw, sel_out, fus_out);

    topk_kernel<<<TOKENS / 256, 256, 0, stream>>>(
        sel_out, topk_vals, topk_idx);

    stats_kernel<<<1, 256, 0, stream>>>(
        topk_idx, usage, new_mw, new_usage);
}